// SparseTransE_47665547051863
// MI455X (gfx1250) — compile-verified
//
#include <hip/hip_runtime.h>

typedef __attribute__((ext_vector_type(2))) float v2f;
typedef __attribute__((ext_vector_type(8))) float v8f;

#define EMB 256
#define TILE 16
#define WAVES_PER_BLOCK 4
#define LROW  34            // padded LDS row stride in floats (conflict-avoiding, 8B-aligned pairs)
#define LPROD (16 * LROW)   // one 16x32 partial matrix (padded)
#define LWAVE (6 * LPROD)   // six products per wave

__device__ __forceinline__ float dot44(float4 a, float4 b, float4 c, float4 d) {
    float s = a.x * b.x;
    s = fmaf(a.y, b.y, s);
    s = fmaf(a.z, b.z, s);
    s = fmaf(a.w, b.w, s);
    s = fmaf(c.x, d.x, s);
    s = fmaf(c.y, d.y, s);
    s = fmaf(c.z, d.z, s);
    s = fmaf(c.w, d.w, s);
    return s;
}

__global__ __launch_bounds__(128) void transe_score_kernel(
    const float* __restrict__ emb,
    const int* __restrict__ ph, const int* __restrict__ pr, const int* __restrict__ pt,
    const int* __restrict__ nh, const int* __restrict__ nr, const int* __restrict__ nt,
    const int* __restrict__ nentp,
    float* __restrict__ out, int Bsz, int T)
{
    __shared__ float lds[WAVES_PER_BLOCK * LWAVE];

    const int lane = threadIdx.x & 31;
    const int wave = threadIdx.x >> 5;
    const int tile = blockIdx.x * WAVES_PER_BLOCK + wave;
    const int tileBase = tile * TILE;
    if (tileBase >= T) return;

    const int n_ent = nentp[0];
    float* P = &lds[wave * LWAVE];

    // ---------------- Phase 1: gather + per-lane partial dot products ----------------
    // lane covers columns [4*lane, 4*lane+3] and [128+4*lane, 128+4*lane+3]
    for (int m = 0; m < TILE; ++m) {
        int tg = tileBase + m;
        if (tg >= T) tg = T - 1;           // defensive clamp (T is a multiple of 16 here)
        int hi, ri, ti;
        if (tg < Bsz) { hi = ph[tg]; ri = pr[tg]; ti = pt[tg]; }
        else          { int u = tg - Bsz; hi = nh[u]; ri = nr[u]; ti = nt[u]; }

        const float4* hp = (const float4*)(emb + (size_t)hi * EMB);
        const float4* rp = (const float4*)(emb + (size_t)(n_ent + ri) * EMB);
        const float4* tp = (const float4*)(emb + (size_t)ti * EMB);

        float4 ha = hp[lane], hb = hp[lane + 32];   // global_load_b128, fully coalesced
        float4 ra = rp[lane], rb = rp[lane + 32];
        float4 ta = tp[lane], tb = tp[lane + 32];

        float hh = dot44(ha, ha, hb, hb);
        float rr = dot44(ra, ra, rb, rb);
        float tt = dot44(ta, ta, tb, tb);
        float hr = dot44(ha, ra, hb, rb);
        float ht = dot44(ha, ta, hb, tb);
        float rt = dot44(ra, ta, rb, tb);

        const int sbase = m * LROW + lane;
        P[0 * LPROD + sbase] = hh;
        P[1 * LPROD + sbase] = rr;
        P[2 * LPROD + sbase] = tt;
        P[3 * LPROD + sbase] = hr;
        P[4 * LPROD + sbase] = ht;
        P[5 * LPROD + sbase] = rt;
    }

    // Per-wave LDS ops are in-order (DScnt); fence the compiler and drain DS.
    asm volatile("s_wait_dscnt 0" ::: "memory");

    // ---------------- Phase 2: WMMA row-sum reduction (D = A x ones + C) ----------------
    // f32 16x16x4 A layout: lanes 0-15 hold K=0,1 ; lanes 16-31 hold K=2,3 for row = lane&15
    const int row  = lane & 15;
    const int koff = (lane >> 4) << 1;

    v2f bones; bones.x = 1.0f; bones.y = 1.0f;   // all-ones B: layout-invariant

    v8f acc[6];
    #pragma unroll
    for (int p = 0; p < 6; ++p) {
        v8f c = {0.f, 0.f, 0.f, 0.f, 0.f, 0.f, 0.f, 0.f};
        const float* abase = &P[p * LPROD + row * LROW + koff];
        #pragma unroll
        for (int s = 0; s < 8; ++s) {
            v2f a;
            a.x = abase[4 * s];
            a.y = abase[4 * s + 1];
            c = __builtin_amdgcn_wmma_f32_16x16x4_f32(
                    /*neg_a=*/false, a, /*neg_b=*/false, bones,
                    /*c_mod=*/(short)0, c, /*reuse_a=*/false, /*reuse_b=*/false);
        }
        acc[p] = c;
    }

    // ---------------- Phase 3: finalize scores, store ----------------
    // D layout: VGPR j holds triple m = j + 8*(lane>=16); columns replicated, use row==0 lanes.
    if (row == 0) {
        float sc[8];
        #pragma unroll
        for (int j = 0; j < 8; ++j) {
            float hh = acc[0][j], rr = acc[1][j], tt = acc[2][j];
            float hr = acc[3][j], ht = acc[4][j], rt = acc[5][j];
            float nhn = fmaxf(sqrtf(hh), 1e-12f);
            float ntn = fmaxf(sqrtf(tt), 1e-12f);
            float ih = 1.0f / nhn;
            float it = 1.0f / ntn;
            float s  = hh * ih * ih + rr + tt * it * it
                     + 2.0f * hr * ih - 2.0f * ht * (ih * it) - 2.0f * rt * it;
            sc[j] = -s;
        }
        float4* o = (float4*)(out + tileBase + ((lane >> 4) << 3));
        o[0] = make_float4(sc[0], sc[1], sc[2], sc[3]);
        o[1] = make_float4(sc[4], sc[5], sc[6], sc[7]);
    }
}

extern "C" void kernel_launch(void* const* d_in, const int* in_sizes, int n_in,
                              void* d_out, int out_size, void* d_ws, size_t ws_size,
                              hipStream_t stream) {
    const float* emb = (const float*)d_in[0];
    const int* ph    = (const int*)d_in[1];
    const int* pr    = (const int*)d_in[2];
    const int* pt    = (const int*)d_in[3];
    const int* nh    = (const int*)d_in[4];
    const int* nr    = (const int*)d_in[5];
    const int* nt    = (const int*)d_in[6];
    const int* nentp = (const int*)d_in[7];
    float* out = (float*)d_out;

    const int Bsz = in_sizes[1];          // 65536
    const int T   = out_size;             // 2*B = 131072 (pos scores then neg scores)

    const int tiles  = (T + TILE - 1) / TILE;
    const int blocks = (tiles + WAVES_PER_BLOCK - 1) / WAVES_PER_BLOCK;

    hipLaunchKernelGGL(transe_score_kernel, dim3(blocks), dim3(WAVES_PER_BLOCK * 32), 0, stream,
                       emb, ph, pr, pt, nh, nr, nt, nentp, out, Bsz, T);
}